// Spatial_Attention_40398462386246
// MI455X (gfx1250) — compile-verified
//
#include <hip/hip_runtime.h>
#include <hip/hip_bf16.h>
#include <math.h>

// ---------------------------------------------------------------------------
// spatial_attention for b=32, t=243, s=17, c=256 (HEAD=8, fd=64, channel=1088)
// All GEMMs use V_WMMA_F32_16X16X4_F32 (full fp32 matrix pipe on gfx1250).
// GEMM inner loop is register double-buffered so WMMA waits target loads
// issued one full K-step earlier. Attention K/V staging uses
// GLOBAL_LOAD_ASYNC_TO_LDS_B32 (ASYNCcnt-tracked) to de-interleave qkv.
// ---------------------------------------------------------------------------

typedef __attribute__((ext_vector_type(2))) float v2f;
typedef __attribute__((ext_vector_type(8))) float v8f;

#define B_  32
#define T_  243
#define S_  17
#define C_  256
#define FD_ 64
#define CH_ 32
#define M1_ (B_*T_*S_)      // 132192 rows over (b,t,s)
#define M2_ (B_*T_)         // 7776  rows over (b,t)
#define CHN_ (S_*FD_)       // 1088

#if __has_builtin(__builtin_amdgcn_global_load_async_to_lds_b32) && \
    __has_builtin(__builtin_amdgcn_s_wait_asynccnt)
#define HAVE_ASYNC_LDS 1
typedef __attribute__((address_space(1))) int* gint_p;   // global (AS1)
typedef __attribute__((address_space(3))) int* lint_p;   // LDS    (AS3)
#else
#define HAVE_ASYNC_LDS 0
#endif

// ---------------------------------------------------------------------------
// Generic fp32 WMMA GEMM:  C = A[M,K] @ B[K,N] + bias (+resid) (+=C if accum)
// Block = 256 threads = 8 waves. Wave computes a 64x16 tile (4 M-subtiles
// sharing one B fragment). Block tile = 64 x 128.
// A fragment (16x4 f32): lane<16 -> row=lane, K=2h..2h+1 with h=lane>>4.
// B fragment (4x16):     VGPR0 = B[2h][col], VGPR1 = B[2h+1][col].
// D (16x16 f32): vgpr r -> row = r + 8*h, col = lane&15.
// ---------------------------------------------------------------------------
__global__ __launch_bounds__(256) void gemm_wmma_f32(
    const float* __restrict__ A, const float* __restrict__ Bm,
    const float* __restrict__ bias, const float* __restrict__ resid,
    float* __restrict__ C, int M, int N, int K, int accum)
{
  const int wave = threadIdx.x >> 5;
  const int lane = threadIdx.x & 31;
  const int half = lane >> 4;          // 0 or 1
  const int l15  = lane & 15;
  const int n0   = blockIdx.x * 128 + wave * 16;
  const int m0   = blockIdx.y * 64;

  const int col  = n0 + l15;
  const int colc = (col < N) ? col : (N - 1);

  int m0r = m0 + 0*16 + l15;  int m1r = m0 + 1*16 + l15;
  int m2r = m0 + 2*16 + l15;  int m3r = m0 + 3*16 + l15;
  const float* a0p = A + (size_t)((m0r < M) ? m0r : M-1) * K + half*2;
  const float* a1p = A + (size_t)((m1r < M) ? m1r : M-1) * K + half*2;
  const float* a2p = A + (size_t)((m2r < M) ? m2r : M-1) * K + half*2;
  const float* a3p = A + (size_t)((m3r < M) ? m3r : M-1) * K + half*2;
  const float* bp  = Bm + (size_t)(half*2) * N + colc;

  v8f acc0 = {}, acc1 = {}, acc2 = {}, acc3 = {};

  // Prologue: fragments for k = 0.
  v2f bf, a0, a1, a2, a3;
  bf.x = bp[0];
  bf.y = bp[N];
  a0 = *(const v2f*)a0p;
  a1 = *(const v2f*)a1p;
  a2 = *(const v2f*)a2p;
  a3 = *(const v2f*)a3p;

  // Pipelined main loop: issue loads for k+4, then WMMA on k.
  for (int k = 0; k + 4 < K; k += 4) {
    __builtin_prefetch(bp + 12*(size_t)N, 0, 3);      // global_prefetch_b8
    const float* bpn = bp + 4*(size_t)N;
    v2f bfn;
    bfn.x = bpn[0];
    bfn.y = bpn[N];
    v2f a0n = *(const v2f*)(a0p + 4);
    v2f a1n = *(const v2f*)(a1p + 4);
    v2f a2n = *(const v2f*)(a2p + 4);
    v2f a3n = *(const v2f*)(a3p + 4);
    acc0 = __builtin_amdgcn_wmma_f32_16x16x4_f32(false, a0, false, bf, (short)0, acc0, false, false);
    acc1 = __builtin_amdgcn_wmma_f32_16x16x4_f32(false, a1, false, bf, (short)0, acc1, false, false);
    acc2 = __builtin_amdgcn_wmma_f32_16x16x4_f32(false, a2, false, bf, (short)0, acc2, false, false);
    acc3 = __builtin_amdgcn_wmma_f32_16x16x4_f32(false, a3, false, bf, (short)0, acc3, false, false);
    bf = bfn; a0 = a0n; a1 = a1n; a2 = a2n; a3 = a3n;
    bp = bpn; a0p += 4; a1p += 4; a2p += 4; a3p += 4;
  }
  // Epilogue WMMA group (last K-step).
  acc0 = __builtin_amdgcn_wmma_f32_16x16x4_f32(false, a0, false, bf, (short)0, acc0, false, false);
  acc1 = __builtin_amdgcn_wmma_f32_16x16x4_f32(false, a1, false, bf, (short)0, acc1, false, false);
  acc2 = __builtin_amdgcn_wmma_f32_16x16x4_f32(false, a2, false, bf, (short)0, acc2, false, false);
  acc3 = __builtin_amdgcn_wmma_f32_16x16x4_f32(false, a3, false, bf, (short)0, acc3, false, false);

  const bool colok = (col < N);
  const float bv = bias ? bias[colc] : 0.0f;

  #pragma unroll
  for (int i = 0; i < 4; ++i) {
    v8f acc = (i == 0) ? acc0 : (i == 1) ? acc1 : (i == 2) ? acc2 : acc3;
    #pragma unroll
    for (int r = 0; r < 8; ++r) {
      int row = m0 + i*16 + half*8 + r;
      if (colok && row < M) {
        size_t o = (size_t)row * N + col;
        float v = acc[r] + bv;
        if (resid) v += resid[o];
        if (accum) v += C[o];
        C[o] = v;
      }
    }
  }
}

// ---------------------------------------------------------------------------
// DCT-II / inverse DCT over the s=17 axis, fd=64 lanes.
// in/out layout: [bt][17][64] flat. One block of 64 threads per (b,t).
// fwd:  out[k][f] = sum_s 2*cos(pi*(2s+1)*k/34) * in[s][f]
// inv:  out[n][f] = sum_k cos(pi*k*(2n+1)/34)/17 * (k==0?0.5:1) * in[k][f]
// ---------------------------------------------------------------------------
__global__ __launch_bounds__(64) void dct17_kernel(
    const float* __restrict__ in, float* __restrict__ out, int inverse)
{
  __shared__ float Dt[17*17];
  const float PI = 3.14159265358979323846f;
  for (int idx = threadIdx.x; idx < 289; idx += 64) {
    int a = idx / 17, b2 = idx % 17;   // fwd: a=k,b2=s ; inv: a=n,b2=k
    float w;
    if (!inverse) {
      w = 2.0f * cosf(PI * (2.0f*b2 + 1.0f) * (float)a / 34.0f);
    } else {
      w = cosf(PI * (float)b2 * (2.0f*a + 1.0f) / 34.0f) / 17.0f;
      if (b2 == 0) w *= 0.5f;
    }
    Dt[idx] = w;
  }
  __syncthreads();

  const size_t base = (size_t)blockIdx.x * (S_ * FD_);
  const int f = threadIdx.x;
  float xv[17];
  #pragma unroll
  for (int s2 = 0; s2 < 17; ++s2) xv[s2] = in[base + s2*FD_ + f];
  #pragma unroll
  for (int k = 0; k < 17; ++k) {
    float acc = 0.0f;
    #pragma unroll
    for (int s2 = 0; s2 < 17; ++s2) acc += Dt[k*17 + s2] * xv[s2];
    out[base + k*FD_ + f] = acc;
  }
}

// Exact GELU (erf form), elementwise.
__global__ __launch_bounds__(256) void gelu_kernel(
    const float* __restrict__ in, float* __restrict__ out, long long n)
{
  long long i = (long long)blockIdx.x * 256 + threadIdx.x;
  long long stride = (long long)gridDim.x * 256;
  for (; i < n; i += stride) {
    float x = in[i];
    out[i] = 0.5f * x * (1.0f + erff(x * 0.7071067811865476f));
  }
}

// ---------------------------------------------------------------------------
// Depthwise 3x3 conv over (t,s) on v (v lives interleaved in qkv at +2,
// stride 3). Output HDPE stored in [b,t,s,c] layout (matches d_out layout).
// ---------------------------------------------------------------------------
__global__ __launch_bounds__(256) void dwconv_kernel(
    const float* __restrict__ qkv, const float* __restrict__ Wc,
    const float* __restrict__ bc, float* __restrict__ H)
{
  const long long TOT = (long long)M1_ * C_;
  long long idx = (long long)blockIdx.x * 256 + threadIdx.x;
  if (idx >= TOT) return;
  int cc = (int)(idx & (C_ - 1));
  long long rest = idx >> 8;
  int s2 = (int)(rest % S_);
  long long bt = rest / S_;
  int t = (int)(bt % T_);
  int b = (int)(bt / T_);

  float acc = bc[cc];
  #pragma unroll
  for (int dt = -1; dt <= 1; ++dt) {
    int tt = t + dt;
    if (tt < 0 || tt >= T_) continue;
    #pragma unroll
    for (int ds = -1; ds <= 1; ++ds) {
      int ss = s2 + ds;
      if (ss < 0 || ss >= S_) continue;
      size_t vi = ((size_t)((b*T_ + tt)*S_ + ss)) * (3*C_) + cc*3 + 2;
      acc += qkv[vi] * Wc[cc*9 + (dt+1)*3 + (ds+1)];
    }
  }
  H[idx] = acc;
}

// ---------------------------------------------------------------------------
// Attention + HDPE + 0.001 * maxpool_t(HDPE).  One block (256 thr) per (b,t).
// 8 heads x 32 lanes; lane i (<17) owns row s=i: computes softmax(q·k^T/16)@v.
// K/V staged into LDS via async global->LDS (de-interleaving the stride-3
// qkv layout with per-lane LDS destination addresses) when available.
// ---------------------------------------------------------------------------
__global__ __launch_bounds__(256) void attn_kernel(
    const float* __restrict__ qkv, const float* __restrict__ H,
    float* __restrict__ out)
{
  __shared__ float ks_[S_ * C_];
  __shared__ float vs [S_ * C_];
  const int tid = threadIdx.x;
  const int bt = blockIdx.x;
  const size_t qbase = (size_t)bt * S_ * (3*C_);

#if HAVE_ASYNC_LDS
  for (int idx = tid; idx < S_ * 3 * C_; idx += 256) {
    int s2 = idx / (3*C_);
    int r  = idx - s2 * (3*C_);
    int cc = r / 3;
    int w  = r - cc*3;
    if (w) {   // EXEC-masked async copies; q elements (w==0) skipped
      float* dst = (w == 1) ? &ks_[s2*C_ + cc] : &vs[s2*C_ + cc];
      __builtin_amdgcn_global_load_async_to_lds_b32(
          (gint_p)(qkv + qbase + idx), (lint_p)dst, 0, 0);
    }
  }
  __builtin_amdgcn_s_wait_asynccnt(0);
  __syncthreads();
#else
  for (int idx = tid; idx < S_ * 3 * C_; idx += 256) {
    float val = qkv[qbase + idx];
    int s2 = idx / (3*C_);
    int r  = idx - s2 * (3*C_);
    int cc = r / 3;
    int w  = r - cc*3;
    if (w == 1)      ks_[s2*C_ + cc] = val;
    else if (w == 2) vs [s2*C_ + cc] = val;
  }
  __syncthreads();
#endif

  const int head = tid >> 5;
  const int lane = tid & 31;
  const int i = (lane < S_) ? lane : (S_ - 1);
  const int hc = head * CH_;

  float q[CH_];
  #pragma unroll
  for (int c2 = 0; c2 < CH_; ++c2)
    q[c2] = qkv[qbase + (size_t)i*(3*C_) + (hc + c2)*3];

  float sc[S_];
  #pragma unroll
  for (int j = 0; j < S_; ++j) {
    float d = 0.0f;
    #pragma unroll
    for (int c2 = 0; c2 < CH_; ++c2) d += q[c2] * ks_[j*C_ + hc + c2];
    sc[j] = d * 0.0625f;                       // scale = c^-0.5 = 1/16
  }
  float mx = sc[0];
  #pragma unroll
  for (int j = 1; j < S_; ++j) mx = fmaxf(mx, sc[j]);
  float sum = 0.0f;
  #pragma unroll
  for (int j = 0; j < S_; ++j) { sc[j] = expf(sc[j] - mx); sum += sc[j]; }
  const float inv = 1.0f / sum;

  float o[CH_];
  #pragma unroll
  for (int c2 = 0; c2 < CH_; ++c2) o[c2] = 0.0f;
  #pragma unroll
  for (int j = 0; j < S_; ++j) {
    float a = sc[j] * inv;
    #pragma unroll
    for (int c2 = 0; c2 < CH_; ++c2) o[c2] += a * vs[j*C_ + hc + c2];
  }

  if (lane < S_) {
    const int t = bt % T_;
    const size_t obase = (size_t)bt * S_ * C_ + (size_t)i * C_ + hc;
    const size_t tstride = (size_t)S_ * C_;
    #pragma unroll
    for (int c2 = 0; c2 < CH_; ++c2) {
      float Hv = H[obase + c2];
      float L = Hv;
      if (t > 0)      L = fmaxf(L, H[obase - tstride + c2]);
      if (t < T_ - 1) L = fmaxf(L, H[obase + tstride + c2]);
      out[obase + c2] = o[c2] + Hv + 0.001f * L;
    }
  }
}

// ---------------------------------------------------------------------------
static inline void launch_gemm(const float* A, const float* Bm, const float* bias,
                               const float* resid, float* C, int M, int N, int K,
                               int accum, hipStream_t s)
{
  dim3 grid((N + 127) / 128, (M + 63) / 64);
  gemm_wmma_f32<<<grid, 256, 0, s>>>(A, Bm, bias, resid, C, M, N, K, accum);
}

extern "C" void kernel_launch(void* const* d_in, const int* in_sizes, int n_in,
                              void* d_out, int out_size, void* d_ws, size_t ws_size,
                              hipStream_t stream) {
  (void)in_sizes; (void)n_in; (void)out_size; (void)ws_size;
  const float* x     = (const float*)d_in[0];
  const float* Wqkv  = (const float*)d_in[1];
  const float* bqkv  = (const float*)d_in[2];
  const float* Wconv = (const float*)d_in[3];
  const float* bconv = (const float*)d_in[4];
  const float* Wfe   = (const float*)d_in[5];
  const float* bfe   = (const float*)d_in[6];
  const float* Wd1   = (const float*)d_in[7];
  const float* bd1   = (const float*)d_in[8];
  const float* Wd2   = (const float*)d_in[9];
  const float* bd2   = (const float*)d_in[10];
  const float* WdN   = (const float*)d_in[11];
  const float* bdN   = (const float*)d_in[12];
  const float* Wu2   = (const float*)d_in[13];
  const float* bu2   = (const float*)d_in[14];
  const float* Wu1   = (const float*)d_in[15];
  const float* bu1   = (const float*)d_in[16];
  const float* Wue   = (const float*)d_in[17];
  const float* bue   = (const float*)d_in[18];
  float* out = (float*)d_out;

  // Workspace carve-up (floats). Total ~686 MB.
  float* ws = (float*)d_ws;
  const size_t SZ_FE   = (size_t)M1_ * FD_;        // 8,460,288  (reused as out1)
  const size_t SZ_FLAT = (size_t)M2_ * CHN_;       // 8,460,288
  const size_t SZ_D1   = (size_t)M2_ * (CHN_/2);   // 4,230,144
  const size_t SZ_D2   = (size_t)M2_ * (CHN_/4);   // 2,115,072
  float* fe   = ws;                 // also holds out1 (iDCT result) later
  float* flat = fe   + SZ_FE;
  float* d1   = flat + SZ_FLAT;
  float* g1   = d1   + SZ_D1;
  float* d2   = g1   + SZ_D1;
  float* g2   = d2   + SZ_D2;
  float* dN   = g2   + SZ_D2;
  float* u2   = dN   + SZ_D2;
  float* u1   = u2   + SZ_D1;
  float* qkvb = u1   + SZ_FLAT;                    // M1 * 768
  float* hdpe = qkvb + (size_t)M1_ * (3*C_);       // M1 * 256

  const int CHN = CHN_, CH2 = CHN_/2, CH4 = CHN_/4;

  // ---- freqsh branch ----
  launch_gemm(x, Wfe, bfe, nullptr, fe, M1_, FD_, C_, 0, stream);        // fe
  dct17_kernel<<<M2_, 64, 0, stream>>>(fe, flat, 0);                     // DCT -> flat
  launch_gemm(flat, Wd1, bd1, nullptr, d1, M2_, CH2, CHN, 0, stream);    // d1
  gelu_kernel<<<4096, 256, 0, stream>>>(d1, g1, (long long)M2_*CH2);
  launch_gemm(g1, Wd2, bd2, nullptr, d2, M2_, CH4, CH2, 0, stream);      // d2
  gelu_kernel<<<4096, 256, 0, stream>>>(d2, g2, (long long)M2_*CH4);
  launch_gemm(g2, WdN, bdN, nullptr, dN, M2_, CH4, CH4, 0, stream);      // dN
  launch_gemm(dN, Wu2, bu2, d1, u2, M2_, CH2, CH4, 0, stream);           // u2 (+d1)
  launch_gemm(u2, Wu1, bu1, flat, u1, M2_, CHN, CH2, 0, stream);         // u1 (+flat)
  dct17_kernel<<<M2_, 64, 0, stream>>>(u1, fe, 1);                       // iDCT -> out1 (in fe)

  // ---- attention branch ----
  launch_gemm(x, Wqkv, bqkv, nullptr, qkvb, M1_, 3*C_, C_, 0, stream);   // qkv
  {
    long long tot = (long long)M1_ * C_;
    dwconv_kernel<<<(unsigned)((tot + 255) / 256), 256, 0, stream>>>(qkvb, Wconv, bconv, hdpe);
  }
  attn_kernel<<<M2_, 256, 0, stream>>>(qkvb, hdpe, out);                 // writes all of d_out

  // ---- final: out += iDCT(out1) @ W_ue + b_ue ----
  launch_gemm(fe, Wue, bue, nullptr, out, M1_, C_, FD_, 1, stream);
}